// MelPCEN_2783138807956
// MI455X (gfx1250) — compile-verified
//
#include <hip/hip_runtime.h>
#include <hip/hip_bf16.h>
#include <math.h>

// ---------------- problem constants ----------------
#define NFFT     512
#define NBINS    257            // NFFT/2 + 1
#define NMELS    80
#define HOPLEN   160
#define BATCH    32
#define WAVELEN  960000
#define TFRAMES  6001           // 1 + (WAVELEN + 2*256 - 512)/160
#define TILE_T   16
#define NTILES   376            // ceil(6001/16)
#define KPAD     260            // NBINS padded to multiple of 4
#define PSTRIDE  261            // LDS row stride (odd vs 64 banks)
#define SEGN     (15 * HOPLEN + NFFT)   // 2912

// PCEN chunked-scan parameters
#define ROWS     (BATCH * NMELS)        // 2560
#define CHLEN    128
#define NCH      47                     // ceil(6001/128)
#define PCEN_A   0.975f
#define PCEN_B   0.025f

typedef __attribute__((ext_vector_type(2))) float v2f;
typedef __attribute__((ext_vector_type(8))) float v8f;

// gfx1250 async global->LDS copy (VGLOBAL op 96, tracked by ASYNCcnt).
__device__ __forceinline__ void async_g2lds_b32(unsigned lds_addr, const float* gptr) {
    asm volatile("global_load_async_to_lds_b32 %0, %1, off"
                 :: "v"(lds_addr), "v"(gptr) : "memory");
}
__device__ __forceinline__ void wait_asynccnt0() {
    asm volatile("s_wait_asynccnt 0x0" ::: "memory");
}

// ---------------------------------------------------------------------------
// Kernel 0: build mel filterbank fb[KPAD][NMELS] (rows >= 257 are zero)
// ---------------------------------------------------------------------------
__global__ void fb_kernel(float* __restrict__ fb) {
    int i = blockIdx.x * blockDim.x + threadIdx.x;
    if (i >= KPAD * NMELS) return;
    int f = i / NMELS;
    int m = i - f * NMELS;
    float val = 0.0f;
    if (f < NBINS) {
        float freq    = (float)f * (8000.0f / 256.0f);      // linspace(0, sr/2, 257)
        float mel_max = 2595.0f * log10f(1.0f + 8000.0f / 700.0f);
        float mp0 = mel_max * (float)(m)     / 81.0f;       // linspace(0, mel_max, 82)
        float mp1 = mel_max * (float)(m + 1) / 81.0f;
        float mp2 = mel_max * (float)(m + 2) / 81.0f;
        float fp0 = 700.0f * (exp10f(mp0 / 2595.0f) - 1.0f);
        float fp1 = 700.0f * (exp10f(mp1 / 2595.0f) - 1.0f);
        float fp2 = 700.0f * (exp10f(mp2 / 2595.0f) - 1.0f);
        float down = (freq - fp0) / (fp1 - fp0);
        float up   = (fp2 - freq) / (fp2 - fp1);
        val = fmaxf(fminf(down, up), 0.0f);
    }
    fb[i] = val;
}

// ---------------------------------------------------------------------------
// Kernel 1: mel spectrogram.  grid = (NTILES, BATCH), block = 256 (8 wave32s).
// Async-copy waveform segment to LDS, per-wave in-place 512-pt DIT FFT with a
// shared twiddle LUT, then waves 0..4 run the 16x260 @ 260x80 projection on
// V_WMMA_F32_16X16X4_F32.  Output mel[b][t][m] (+1e-9 from exp(log(x+1e-9))).
// ---------------------------------------------------------------------------
__global__ void __launch_bounds__(256)
melspec_kernel(const float* __restrict__ wave,
               const float* __restrict__ fb,
               float* __restrict__ mel) {
    __shared__ float  seg[SEGN];            // 11648 B
    __shared__ float2 fbuf[8 * NFFT];       // 32768 B: one 512-pt buffer per wave
    __shared__ float  P[TILE_T * PSTRIDE];  // 16704 B
    __shared__ float2 tw[256];              //  2048 B: exp(-2*pi*i*k/512)

    const int tile = blockIdx.x;
    const int b    = blockIdx.y;
    const int t0   = tile * TILE_T;
    const int tid  = threadIdx.x;
    const int lane = tid & 31;
    const int wv   = tid >> 5;

    // ---- async-load waveform segment with reflect padding (pad = 256) ----
    const float* wrow = wave + (size_t)b * WAVELEN;
    const long   base = (long)t0 * HOPLEN - 256;
    for (int i = tid; i < SEGN; i += 256) {
        long p = base + (long)i;
        long q = (p < 0) ? -p : ((p >= WAVELEN) ? (2L * WAVELEN - 2 - p) : p);
        async_g2lds_b32((unsigned)(uintptr_t)&seg[i], wrow + q);
    }
    // twiddle LUT (256 threads -> one entry each), overlaps with async copy
    {
        float sn, cs;
        __sincosf(-6.283185307179586f * (float)tid * (1.0f / 512.0f), &sn, &cs);
        tw[tid] = make_float2(cs, sn);
    }
    wait_asynccnt0();
    __syncthreads();

    // ---- per-wave: window + 512-pt in-place radix-2 DIT FFT x 2 frames ----
    float2* buf = fbuf + wv * NFFT;
    for (int rep = 0; rep < 2; ++rep) {
        const int fl = wv * 2 + rep;            // local frame 0..15
        const int t  = t0 + fl;
        if (t < TFRAMES) {
            // Hann window (periodic), loaded directly in bit-reversed order
            for (int i = lane; i < NFFT; i += 32) {
                float w = 0.5f - 0.5f * __cosf((float)i * 0.012271846303085130f);
                int j = (int)(__brev((unsigned)i) >> 23);
                buf[j] = make_float2(seg[fl * HOPLEN + i] * w, 0.0f);
            }
            __builtin_amdgcn_wave_barrier();
            int sh = 0;                          // log2(half)
            #pragma unroll
            for (int len = 2; len <= NFFT; len <<= 1) {
                const int half = len >> 1;
                for (int bt = lane; bt < NFFT / 2; bt += 32) {
                    int k  = bt & (half - 1);
                    int g  = bt >> sh;
                    int i0 = (g << (sh + 1)) + k;
                    int i1 = i0 + half;
                    float2 w = tw[k << (8 - sh)];   // exp(-2*pi*i*k/len)
                    float2 u = buf[i0];
                    float2 v = buf[i1];
                    float tr = v.x * w.x - v.y * w.y;
                    float ti = v.x * w.y + v.y * w.x;
                    buf[i0] = make_float2(u.x + tr, u.y + ti);
                    buf[i1] = make_float2(u.x - tr, u.y - ti);
                }
                __builtin_amdgcn_wave_barrier();
                ++sh;
            }
            for (int k = lane; k < NBINS; k += 32) {
                float2 X = buf[k];
                P[fl * PSTRIDE + k] = X.x * X.x + X.y * X.y;
            }
            for (int k = NBINS + lane; k < KPAD; k += 32)
                P[fl * PSTRIDE + k] = 0.0f;
        } else {
            for (int k = lane; k < KPAD; k += 32)
                P[fl * PSTRIDE + k] = 0.0f;
        }
    }
    __syncthreads();

    // ---- mel projection: D(16x16) += A(16x4,f32) x B(4x16,f32), 65 k-steps ----
    if (wv < 5) {                               // uniform branch -> EXEC all 1s
        const int col  = lane & 15;
        const int koff = (lane >> 4) * 2;
        const int mcol = wv * 16 + col;
        v8f acc = {};
        for (int k = 0; k < KPAD; k += 4) {
            v2f a, bb;
            a[0]  = P[col * PSTRIDE + k + koff];
            a[1]  = P[col * PSTRIDE + k + koff + 1];
            bb[0] = fb[(k + koff)     * NMELS + mcol];
            bb[1] = fb[(k + koff + 1) * NMELS + mcol];
            acc = __builtin_amdgcn_wmma_f32_16x16x4_f32(
                      false, a, false, bb, (short)0, acc, false, false);
        }
        const int rbase = (lane >> 4) * 8;
        float* mp = mel + ((size_t)b * TFRAMES + t0) * NMELS + mcol;
        if (t0 + TILE_T <= TFRAMES) {           // full tile: unguarded stores
            #pragma unroll
            for (int j = 0; j < 8; ++j)
                mp[(size_t)(rbase + j) * NMELS] = acc[j] + 1e-9f;
        } else {
            #pragma unroll
            for (int j = 0; j < 8; ++j)
                if (t0 + rbase + j < TFRAMES)
                    mp[(size_t)(rbase + j) * NMELS] = acc[j] + 1e-9f;
        }
    }
}

// ---------------------------------------------------------------------------
// PCEN as a chunked affine scan:  s[t] = a*s[t-1] + b*x[t],  s[0] = x[0].
// K2a: per-(row,chunk) local carry.  K2b: per-row combine across chunks.
// K2c: per-(row,chunk) recompute + pointwise output (fully parallel transcend).
// ---------------------------------------------------------------------------
__global__ void pcen_carry_kernel(const float* __restrict__ mel,
                                  float* __restrict__ carry) {
    const int row = blockIdx.x * blockDim.x + threadIdx.x;   // 0..2559
    const int c   = blockIdx.y;                              // 0..NCH-1
    const int bb  = row / NMELS;
    const int m   = row - bb * NMELS;
    const float* x = mel + (size_t)bb * TFRAMES * NMELS + m;
    const int ts = c * CHLEN;
    const int te = (ts + CHLEN < TFRAMES) ? ts + CHLEN : TFRAMES;
    float s;
    if (c == 0) {
        s = x[0];
        for (int t = 1; t < te; ++t)
            s = PCEN_A * s + PCEN_B * x[(size_t)t * NMELS];
    } else {
        s = 0.0f;
        for (int t = ts; t < te; ++t)
            s = PCEN_A * s + PCEN_B * x[(size_t)t * NMELS];
    }
    carry[(size_t)c * ROWS + row] = s;
}

__global__ void pcen_combine_kernel(const float* __restrict__ carry,
                                    float* __restrict__ sin_) {
    const int row = blockIdx.x * blockDim.x + threadIdx.x;
    // a^128 by repeated squaring (deterministic)
    float a2 = PCEN_A * PCEN_A, a4 = a2 * a2, a8 = a4 * a4, a16 = a8 * a8;
    float a32 = a16 * a16, a64 = a32 * a32, a128 = a64 * a64;
    float S = carry[row];                       // true state at end of chunk 0
    sin_[row] = 0.0f;                           // chunk 0 has no incoming state
    for (int c = 1; c < NCH; ++c) {
        sin_[(size_t)c * ROWS + row] = S;
        S = carry[(size_t)c * ROWS + row] + a128 * S;
    }
}

__global__ void pcen_emit_kernel(const float* __restrict__ mel,
                                 const float* __restrict__ sin_,
                                 float* __restrict__ out) {
    const int row = blockIdx.x * blockDim.x + threadIdx.x;   // 0..2559
    const int c   = blockIdx.y;
    const int bb  = row / NMELS;
    const int m   = row - bb * NMELS;
    const float* x = mel + (size_t)bb * TFRAMES * NMELS + m;
    float* o = out + ((size_t)bb * NMELS + m) * TFRAMES;
    const int ts = c * CHLEN;
    const int te = (ts + CHLEN < TFRAMES) ? ts + CHLEN : TFRAMES;
    const float sq2 = 1.4142135623730951f;
    if (c == 0) {
        float s = x[0];
        for (int t = 0; t < te; ++t) {
            float xt = x[(size_t)t * NMELS];
            if (t > 0) s = PCEN_A * s + PCEN_B * xt;
            float v = xt * __powf(s + 1e-6f, -0.98f) + 2.0f;
            o[t] = sqrtf(v) - sq2;
        }
    } else {
        const float S = sin_[(size_t)c * ROWS + row];
        float l = 0.0f, p = 1.0f;
        for (int t = ts; t < te; ++t) {
            float xt = x[(size_t)t * NMELS];
            l = PCEN_A * l + PCEN_B * xt;
            p *= PCEN_A;
            float s = fmaf(p, S, l);
            float v = xt * __powf(s + 1e-6f, -0.98f) + 2.0f;
            o[t] = sqrtf(v) - sq2;
        }
    }
}

// ---------------------------------------------------------------------------
extern "C" void kernel_launch(void* const* d_in, const int* in_sizes, int n_in,
                              void* d_out, int out_size, void* d_ws, size_t ws_size,
                              hipStream_t stream) {
    const float* wave = (const float*)d_in[0];
    float* out = (float*)d_out;

    // ws layout
    size_t offMel = 256 * 1024;                                   // after fb
    size_t melBytes = (size_t)BATCH * TFRAMES * NMELS * sizeof(float);
    size_t offCar = (offMel + melBytes + 255) & ~(size_t)255;
    size_t carBytes = (size_t)NCH * ROWS * sizeof(float);
    size_t offSin = (offCar + carBytes + 255) & ~(size_t)255;

    float* fb    = (float*)d_ws;
    float* mel   = (float*)((char*)d_ws + offMel);
    float* carry = (float*)((char*)d_ws + offCar);
    float* sin_  = (float*)((char*)d_ws + offSin);

    fb_kernel<<<(KPAD * NMELS + 255) / 256, 256, 0, stream>>>(fb);

    dim3 g1(NTILES, BATCH);
    melspec_kernel<<<g1, 256, 0, stream>>>(wave, fb, mel);

    dim3 g2(ROWS / 256, NCH);
    pcen_carry_kernel<<<g2, 256, 0, stream>>>(mel, carry);
    pcen_combine_kernel<<<ROWS / 256, 256, 0, stream>>>(carry, sin_);
    pcen_emit_kernel<<<g2, 256, 0, stream>>>(mel, sin_, out);
}